// SDFGrid_42674795053876
// MI455X (gfx1250) — compile-verified
//
#include <hip/hip_runtime.h>
#include <math.h>

#define NGRID 128
#define NRAYS 16384
#define MAX_STEPS (3 * NGRID)

__device__ __forceinline__ float lerpf(float u, float a, float b) {
    return a + u * (b - a);
}

__global__ __launch_bounds__(128) void sdf_trace_kernel(
    const float* __restrict__ origins,
    const float* __restrict__ dirs,
    const float* __restrict__ sdf,
    float* __restrict__ out_point,
    float* __restrict__ out_normal,
    float* __restrict__ out_hit)
{
    const int i = blockIdx.x * blockDim.x + threadIdx.x;
    if (i >= NRAYS) return;

    const float minb  = -1.0f;
    const float maxb  =  1.0f;
    const float voxel = 2.0f / 127.0f;   // (MAXB-MINB)/(N-1) in f32

    const float ox = origins[3 * i + 0];
    const float oy = origins[3 * i + 1];
    const float oz = origins[3 * i + 2];
    const float dx = dirs[3 * i + 0];
    const float dy = dirs[3 * i + 1];
    const float dz = dirs[3 * i + 2];

    // ---- slab test (matches reference exactly) ----
    const float invx = 1.0f / dx, invy = 1.0f / dy, invz = 1.0f / dz;
    const float t1x = (minb - ox) * invx, t2x = (maxb - ox) * invx;
    const float t1y = (minb - oy) * invy, t2y = (maxb - oy) * invy;
    const float t1z = (minb - oz) * invz, t2z = (maxb - oz) * invz;
    float t_near = fmaxf(fminf(t1x, t2x), fmaxf(fminf(t1y, t2y), fminf(t1z, t2z)));
    const float t_far = fminf(fmaxf(t1x, t2x), fminf(fmaxf(t1y, t2y), fmaxf(t1z, t2z)));
    t_near = (t_near > 0.0f) ? t_near : 0.0f;
    const bool valid = (t_near <= t_far);
    if (!valid) t_near = 0.0f;

    float px = ox + t_near * dx;
    float py = oy + t_near * dy;
    float pz = oz + t_near * dz;

    int ix = (int)floorf((px - minb) / voxel);
    int iy = (int)floorf((py - minb) / voxel);
    int iz = (int)floorf((pz - minb) / voxel);

    const int sx = (dx > 0.0f) ? 1 : ((dx < 0.0f) ? -1 : 0);
    const int sy = (dy > 0.0f) ? 1 : ((dy < 0.0f) ? -1 : 0);
    const int sz = (dz > 0.0f) ? 1 : ((dz < 0.0f) ? -1 : 0);

    const float tdx = voxel / fabsf(dx);
    const float tdy = voxel / fabsf(dy);
    const float tdz = voxel / fabsf(dz);

    float tnextx = (((float)(ix + (sx > 0 ? 1 : 0))) * voxel + minb - ox) / dx;
    float tnexty = (((float)(iy + (sy > 0 ? 1 : 0))) * voxel + minb - oy) / dy;
    float tnextz = (((float)(iz + (sz > 0 ? 1 : 0))) * voxel + minb - oz) / dz;

    bool hit = false;
    float hpx = 0.0f, hpy = 0.0f, hpz = 0.0f;
    float hnx = 0.0f, hny = 0.0f, hnz = 0.0f;

    if (valid) {
        for (int it = 0; it < MAX_STEPS; ++it) {
            // DDA integer-index range check
            if (!(ix >= 0 && ix < 127 && iy >= 0 && iy < 127 && iz >= 0 && iz < 127))
                break;

            const int qx = (int)floorf((px - minb) / voxel);
            const int qy = (int)floorf((py - minb) / voxel);
            const int qz = (int)floorf((pz - minb) / voxel);
            const bool qvalid = (qx >= 0 && qx < 127 && qy >= 0 && qy < 127 &&
                                 qz >= 0 && qz < 127);
            const int cx = min(max(qx, 0), 126);
            const int cy = min(max(qy, 0), 126);
            const int cz = min(max(qz, 0), 126);

            const float* base = sdf + (((size_t)cx * NGRID + (size_t)cy) * NGRID + (size_t)cz);
            // corners: s[a][b][c] at base + a*N*N + b*N + c
            const float s000 = base[0];
            const float s001 = base[1];
            const float s010 = base[NGRID];
            const float s011 = base[NGRID + 1];
            const float s100 = base[NGRID * NGRID];
            const float s101 = base[NGRID * NGRID + 1];
            const float s110 = base[NGRID * NGRID + NGRID];
            const float s111 = base[NGRID * NGRID + NGRID + 1];

            // ---- gfx1250 prefetch of predicted next DDA cell, overlapped
            // with the cubic solve below (global_prefetch_b8) ----
            {
                const float pmin = fminf(tnextx, fminf(tnexty, tnextz));
                int nxp = ix + ((tnextx == pmin) ? sx : 0);
                int nyp = iy + ((tnexty == pmin) ? sy : 0);
                int nzp = iz + ((tnextz == pmin) ? sz : 0);
                nxp = min(max(nxp, 0), 126);
                nyp = min(max(nyp, 0), 126);
                nzp = min(max(nzp, 0), 126);
                const float* nb = sdf + (((size_t)nxp * NGRID + (size_t)nyp) * NGRID + (size_t)nzp);
                __builtin_prefetch(nb, 0, 3);                  // x-plane 0 cacheline
                __builtin_prefetch(nb + NGRID * NGRID, 0, 3);  // x-plane 1 cacheline
            }

            const float p0x = minb + (float)cx * voxel;
            const float p0y = minb + (float)cy * voxel;
            const float p0z = minb + (float)cz * voxel;

            const float Ox = (ox - p0x) / voxel;
            const float Oy = (oy - p0y) / voxel;
            const float Oz = (oz - p0z) / voxel;
            const float Dx = dx / voxel;
            const float Dy = dy / voxel;
            const float Dz = dz / voxel;

            const float k0 = s000;
            const float k1 = s100 - s000;
            const float k2 = s010 - s000;
            const float k3 = s110 - s010 - k1;
            const float k4 = k0 - s001;
            const float av = s101 - s001;
            const float k5 = k1 - av;
            const float k6 = k2 - (s011 - s001);
            const float k7 = k3 - (s111 - s011 - av);

            const float m0 = Ox * Oy;
            const float m1 = Dx * Dy;
            const float m2 = Ox * Dy + Oy * Dx;
            const float m3 = k5 * Oz - k1;
            const float m4 = k6 * Oz - k2;
            const float m5 = k7 * Oz - k3;

            const float c0 = k4 * Oz - k0 + Ox * m3 + Oy * m4 + m0 * m5;
            const float c1 = Dx * m3 + Dy * m4 + m2 * m5 +
                             Dz * (k4 + k5 * Ox + k6 * Oy + k7 * m0);
            const float c2 = m1 * m5 + Dz * (k5 * Dx + k6 * Dy + k7 * m2);
            const float c3 = k7 * m1 * Dz;

            // ---- _min_real_root, branch-for-branch ----
            const bool  lin_has  = (c1 > 1e-19f);
            const float lin_root = -c0 / (lin_has ? c1 : 1.0f);
            const bool  use_lin  = (c2 <= 1e-4f);
            const float disc     = c1 * c1 - 4.0f * c2 * c0;
            const bool  quad_has = (disc > 0.0f);
            const float sq       = sqrtf(quad_has ? disc : 1.0f);
            const float c2s      = use_lin ? 1.0f : c2;
            const float quad_root = fminf((-c1 + sq) / (2.0f * c2s),
                                          (-c1 - sq) / (2.0f * c2s));
            const float q_root = use_lin ? lin_root : quad_root;
            const bool  q_has  = use_lin ? lin_has : quad_has;

            const bool  use_quad = (c3 <= 1e-4f);
            const float c3s = use_quad ? 1.0f : c3;
            const float a  = c2 / c3s;
            const float b  = c1 / c3s;
            const float cc = c0 / c3s;
            const float Q  = (a * a - 3.0f * b) / 9.0f;
            const float R  = (2.0f * a * a * a - 9.0f * a * b + 27.0f * cc) / 54.0f;
            const float Q3 = Q * Q * Q;
            const bool  trig = (R * R < Q3);
            const float Qs   = trig ? Q : 1.0f;
            const float theta = acosf(fminf(fmaxf(R / sqrtf(Qs * Qs * Qs), -1.0f), 1.0f));
            const float t_trig = -2.0f * sqrtf(Qs) * cosf(theta / 3.0f) - a / 3.0f;
            const float inner = trig ? 1.0f : fmaxf(R * R - Q3, 0.0f);
            const float signR = (R > 0.0f) ? 1.0f : ((R < 0.0f) ? -1.0f : 0.0f);
            const float A  = -signR * powf(fabsf(R) + sqrtf(inner), 1.0f / 3.0f);
            const float Bv = (A == 0.0f) ? 0.0f : (Q / A);
            const float t_cub = trig ? t_trig : (A + Bv - a / 3.0f);

            const float t_root = use_quad ? q_root : t_cub;
            const bool  has    = use_quad ? q_has : true;

            const bool hit_now = qvalid && has;
            if (hit_now) {
                const float ipx = ox + t_root * dx;
                const float ipy = oy + t_root * dy;
                const float ipz = oz + t_root * dz;
                const float X = (ipx - p0x) / voxel;
                const float Y = (ipy - p0y) / voxel;
                const float Z = (ipz - p0z) / voxel;
                hnx = lerpf(Z, lerpf(Y, s100 - s000, s110 - s010),
                               lerpf(Y, s101 - s001, s111 - s011));
                hny = lerpf(Z, lerpf(X, s010 - s000, s110 - s100),
                               lerpf(X, s011 - s001, s111 - s101));
                hnz = lerpf(Y, lerpf(X, s001 - s000, s101 - s100),
                               lerpf(X, s011 - s010, s111 - s110));
                hpx = ipx; hpy = ipy; hpz = ipz;
                hit = true;
                break;
            }

            // ---- DDA step ----
            const float tmin = fminf(tnextx, fminf(tnexty, tnextz));
            const bool smx = (tnextx < t_far) && (tnextx == tmin);
            const bool smy = (tnexty < t_far) && (tnexty == tmin);
            const bool smz = (tnextz < t_far) && (tnextz == tmin);
            if (!(smx || smy || smz)) break;   // done: no step possible

            if (smx) { ix += sx; }
            if (smy) { iy += sy; }
            if (smz) { iz += sz; }
            px = ox + tmin * dx;
            py = oy + tmin * dy;
            pz = oz + tmin * dz;
            if (smx) { tnextx += tdx; }
            if (smy) { tnexty += tdy; }
            if (smz) { tnextz += tdz; }
        }
    }

    out_point[3 * i + 0]  = hpx;
    out_point[3 * i + 1]  = hpy;
    out_point[3 * i + 2]  = hpz;
    out_normal[3 * i + 0] = hnx;
    out_normal[3 * i + 1] = hny;
    out_normal[3 * i + 2] = hnz;
    out_hit[i] = hit ? 1.0f : 0.0f;
}

extern "C" void kernel_launch(void* const* d_in, const int* in_sizes, int n_in,
                              void* d_out, int out_size, void* d_ws, size_t ws_size,
                              hipStream_t stream) {
    (void)in_sizes; (void)n_in; (void)d_ws; (void)ws_size; (void)out_size;
    const float* origins = (const float*)d_in[0];
    const float* dirs    = (const float*)d_in[1];
    const float* sdf     = (const float*)d_in[2];

    float* out_point  = (float*)d_out;                 // NRAYS*3
    float* out_normal = out_point + 3 * NRAYS;         // NRAYS*3
    float* out_hit    = out_point + 6 * NRAYS;         // NRAYS

    const int block = 128;
    const int grid  = (NRAYS + block - 1) / block;     // 128 blocks
    sdf_trace_kernel<<<grid, block, 0, stream>>>(origins, dirs, sdf,
                                                 out_point, out_normal, out_hit);
}